// EquivariantSpatialConv_51926154609216
// MI455X (gfx1250) — compile-verified
//
#include <hip/hip_runtime.h>

// ---------------- vector types ----------------
typedef __attribute__((ext_vector_type(2)))  float        v2f;
typedef __attribute__((ext_vector_type(8)))  float        v8f;
typedef __attribute__((ext_vector_type(4)))  unsigned int v4u;
typedef __attribute__((ext_vector_type(8)))  int          v8i_t;
typedef __attribute__((ext_vector_type(4)))  int          v4i_t;

// =====================================================================
// Device-side Wigner-3j construction (runs once per launch, tiny cost)
// =====================================================================
__device__ __forceinline__ double dfac(int n) {
    double r = 1.0;
    for (int i = 2; i <= n; ++i) r *= (double)i;
    return r;
}

__device__ double su2_cg_dev(int j1, int j2, int j3, int m1, int m2, int m3) {
    if (m1 + m2 != m3) return 0.0;
    int vmin = max(max(-j1 + j2 + m3, -j1 + m1), 0);
    int vmax = min(min(j2 + j3 + m1, j3 - j1 + j2), j3 + m3);
    double C = sqrt((double)(2 * j3 + 1) *
        (dfac(j3 + j1 - j2) * dfac(j3 - j1 + j2) * dfac(j1 + j2 - j3) *
         dfac(j3 + m3) * dfac(j3 - m3)) /
        (dfac(j1 + j2 + j3 + 1) * dfac(j1 - m1) * dfac(j1 + m1) *
         dfac(j2 - m2) * dfac(j2 + m2)));
    double S = 0.0;
    for (int v = vmin; v <= vmax; ++v) {
        double t = (dfac(j2 + j3 + m1 - v) * dfac(j1 - m1 + v)) /
                   (dfac(v) * dfac(j3 - j1 + j2 - v) * dfac(j3 + m3 - v) *
                    dfac(v + j1 - j2 - m3));
        S += ((v + j2 + m2) & 1) ? -t : t;
    }
    return C * S;
}

// q matrix element (real->complex SH change of basis, with (-i)^l phase)
__device__ void qval(int l, int r, int c, double& re, double& im) {
    int m = r - l;
    double vr = 0.0, vi = 0.0;
    const double s2 = 0.70710678118654752440;
    if (m < 0) {
        if (c == l - m)      vr =  s2;
        else if (c == l + m) vi = -s2;
    } else if (m == 0) {
        if (c == l) vr = 1.0;
    } else {
        double sg = (m & 1) ? -1.0 : 1.0;
        if (c == l + m)      vr = sg * s2;
        else if (c == l - m) vi = sg * s2;
    }
    switch (l & 3) {               // multiply by (-i)^l
        case 0:  re =  vr; im =  vi; break;
        case 1:  re =  vi; im = -vr; break;
        case 2:  re = -vr; im = -vi; break;
        default: re = -vi; im =  vr; break;
    }
}

__constant__ int c_L1[8] = {4,4,4,4,6,6,6,6};
__constant__ int c_L2[8] = {4,4,6,6,4,4,6,6};
__constant__ int c_L3[8] = {4,6,4,6,4,6,4,6};
__constant__ int c_OFF[9] = {0,729,1782,2835,4356,5409,6930,8451,10648};

__global__ void esc_zero(float* norm) {
    if (threadIdx.x < 8) norm[threadIdx.x] = 0.0f;
}

// One thread per element of each path's (2l1+1)x(2l2+1)x(2l3+1) tensor.
__global__ void esc_wigner(float* base, float* norm) {
    int idx = blockIdx.x * blockDim.x + threadIdx.x;
    if (idx >= 10648) return;
    int path = 0;
    while (idx >= c_OFF[path + 1]) ++path;
    int e  = idx - c_OFF[path];
    int l1 = c_L1[path], l2 = c_L2[path], l3 = c_L3[path];
    int d2 = 2 * l2 + 1, d3 = 2 * l3 + 1;
    int jj = e / (d2 * d3);
    int LL = (e / d3) % d2;
    int mm = e % d3;
    double cr = 0.0;
    for (int i = 0; i < 2 * l1 + 1; ++i) {
        double q1r, q1i; qval(l1, i, jj, q1r, q1i);
        if (q1r == 0.0 && q1i == 0.0) continue;
        for (int k = 0; k < d2; ++k) {
            int n = i + k - l1 - l2 + l3;     // m3 = m1 + m2
            if (n < 0 || n >= d3) continue;
            double cg = su2_cg_dev(l1, l2, l3, i - l1, k - l2, n - l3);
            if (cg == 0.0) continue;
            double q2r, q2i; qval(l2, k, LL, q2r, q2i);
            double q3r, q3i; qval(l3, n, mm, q3r, q3i);
            q3i = -q3i;                        // conjugate
            double ar = q1r * q2r - q1i * q2i;
            double ai = q1r * q2i + q1i * q2r;
            double pr = ar * q3r - ai * q3i;   // real part only (imag cancels)
            cr += pr * cg;
        }
    }
    base[idx] = (float)cr;
    atomicAdd(&norm[path], (float)(cr * cr));
}

// Combine the 8 path tensors + runtime tp_weights into W^T[32][488].
// Row = output channel k (padded to 32). Column = permuted K index kc:
//   s = kc/4, d = kc%4, q = 2s + (d&1), i = q/11,
//   j = (d>=2 ? 11 : 0) + q%11          (lo lane-half owns d=0,1; hi owns d=2,3)
// so both WMMA lane halves share the same compile-time x/y indices.
__global__ void esc_combine(const float* __restrict__ tpw,
                            const float* __restrict__ base,
                            const float* __restrict__ norm,
                            float* __restrict__ Wt) {
    int idx = blockIdx.x * blockDim.x + threadIdx.x;
    if (idx >= 32 * 488) return;
    int k = idx / 488, kc = idx % 488;
    float val = 0.0f;
    if (kc < 484 && k < 22) {
        int s = kc / 4, d = kc % 4;
        int q = 2 * s + (d & 1);
        int i = q / 11;
        int j = (d >= 2 ? 11 : 0) + q % 11;
        int b1 = (i >= 9), b2 = (j >= 9), b3 = (k >= 9);
        int path = b1 * 4 + b2 * 2 + b3;
        int i0 = i - (b1 ? 9 : 0), j0 = j - (b2 ? 9 : 0), k0 = k - (b3 ? 9 : 0);
        int d2 = b2 ? 13 : 9, d3 = b3 ? 13 : 9;
        float pw = b3 ? 1.8027756377319946f /*sqrt(13)/2*/ : 1.5f;
        float inv = rsqrtf(norm[path]);
        val = pw * tpw[path] * inv * base[c_OFF[path] + (i0 * d2 + j0) * d3 + k0];
    }
    Wt[k * 488 + kc] = val;
}

// =====================================================================
// Main fused kernel: depthwise 3x3 conv + (x (x) y) . W GEMM via WMMA
// =====================================================================
#define WAVES_PER_WG 8
#define NWCOL 488
#define NWELEM (32 * NWCOL)   // 15616 floats = 62464 bytes

__global__ __launch_bounds__(256) void esc_main(
    const float* __restrict__ f4, const float* __restrict__ f6,
    const float* __restrict__ sw, const int* __restrict__ pH,
    const int* __restrict__ pW, const float* __restrict__ Wt,
    float* __restrict__ out, int Npix, int nTiles)
{
    __shared__ float sWt[NWELEM];                 // W^T staged by TDM
    __shared__ float sPatch[WAVES_PER_WG][3][18][22];
    __shared__ float sXY[WAVES_PER_WG][16][44];   // [pixel][x(22) | y(22)]

    const int lane   = threadIdx.x & 31;
    const int wave   = threadIdx.x >> 5;
    const int laneLo = lane & 15;
    const bool hiHalf = lane >= 16;

    // ---- TDM: async tensor load of W^T (62464B) into LDS, wave 0 only ----
    if (wave == 0) {
        unsigned long long ga = (unsigned long long)(size_t)Wt;
        v4u g0;
        g0.x = 1u;                                         // count=1, user mode
        g0.y = (unsigned)(size_t)(void*)&sWt[0];           // lds_addr
        g0.z = (unsigned)ga;                               // global_addr[31:0]
        g0.w = (unsigned)((ga >> 32) & 0x01FFFFFFull) | (2u << 30);  // type=2
        const unsigned NW = (unsigned)NWELEM;
        v8i_t g1;
        g1[0] = (int)(2u << 16);                           // data_size = 4B
        g1[1] = (int)((NW & 0xFFFFu) << 16);               // tensor_dim0 lo16
        g1[2] = (int)((NW >> 16) | (1u << 16));            // dim0 hi | tensor_dim1=1
        g1[3] = (int)((NW & 0xFFFFu) << 16);               // tile_dim0
        g1[4] = 1;                                         // tile_dim1 = 1
        g1[5] = (int)NW;                                   // tensor_dim0_stride
        g1[6] = 0;
        g1[7] = 0;
        v4i_t gz  = {0, 0, 0, 0};
        v8i_t gz8 = {0, 0, 0, 0, 0, 0, 0, 0};
        __builtin_amdgcn_tensor_load_to_lds(g0, g1, gz, gz, gz8, 0);
        __builtin_amdgcn_s_wait_tensorcnt(0);
    }

    const int tile   = blockIdx.x * WAVES_PER_WG + wave;
    const bool active = tile < nTiles;
    const int Hh = *pH;
    const int Ww = *pW;
    const int HW = Hh * Ww;

    int h = 0, w0 = 0, b = 0;
    if (active) {
        int pbase = tile * 16;                 // 16 consecutive pixels, one row
        b = pbase / HW;
        int rem = pbase - b * HW;
        h  = rem / Ww;
        w0 = rem - h * Ww;
        // stage 3x18 halo patch (edge clamped), 22 channels per pixel
        for (int t = lane; t < 54; t += 32) {
            int r  = t / 18, c = t % 18;
            int hh = min(max(h + r - 1, 0), Hh - 1);
            int ww = min(max(w0 + c - 1, 0), Ww - 1);
            size_t g = (size_t)(b * Hh + hh) * (size_t)Ww + (size_t)ww;
            const float* p4 = f4 + g * 9;
            const float* p6 = f6 + g * 13;
            #pragma unroll
            for (int ch = 0; ch < 9; ++ch)  sPatch[wave][r][c][ch]     = p4[ch];
            #pragma unroll
            for (int ch = 0; ch < 13; ++ch) sPatch[wave][r][c][9 + ch] = p6[ch];
        }
    }
    __syncthreads();

    // ---- depthwise 3x3 conv -> y, center -> x (lane pair splits channels) ----
    if (active) {
        float swv[9];
        #pragma unroll
        for (int i = 0; i < 9; ++i) swv[i] = sw[i];
        int pix = laneLo;
        int c0  = hiHalf ? 11 : 0;
        #pragma unroll
        for (int ci = 0; ci < 11; ++ci) {
            int ch = c0 + ci;
            float acc = 0.0f;
            #pragma unroll
            for (int r = 0; r < 3; ++r)
                #pragma unroll
                for (int c = 0; c < 3; ++c)
                    acc += swv[r * 3 + c] * sPatch[wave][r][pix + c][ch];
            sXY[wave][pix][ch]      = sPatch[wave][1][pix + 1][ch];   // x
            sXY[wave][pix][22 + ch] = acc;                            // y
        }
    }
    __syncthreads();

    // ---- per-pixel GEMM: out = (x (x) y) . W, K=484 in 121 k-steps ----
    v8f acc0 = {}; v8f acc1 = {};
    if (active) {
        float xv[22], yv[22];
        #pragma unroll
        for (int ch = 0; ch < 22; ++ch) {
            xv[ch] = sXY[wave][laneLo][ch];
            yv[ch] = sXY[wave][laneLo][22 + ch];
        }
        // lane-half-local y window: lo lanes own j = 0..10, hi lanes j = 11..21
        float yy[11];
        #pragma unroll
        for (int c = 0; c < 11; ++c) yy[c] = hiHalf ? yv[11 + c] : yv[c];

        const float* wt0 = &sWt[laneLo * NWCOL];          // N-tile 0 (k=0..15)
        const float* wt1 = &sWt[(16 + laneLo) * NWCOL];   // N-tile 1 (k=16..21)
        #pragma unroll
        for (int s = 0; s < 121; ++s) {
            const int q0 = 2 * s, q1 = 2 * s + 1;
            // A fragment: identical compile-time indices for both lane halves
            v2f a;
            a.x = xv[q0 / 11] * yy[q0 % 11];
            a.y = xv[q1 / 11] * yy[q1 % 11];
            const int ko = 4 * s + (hiHalf ? 2 : 0);
            v2f b0 = *(const v2f*)(wt0 + ko);
            v2f b1 = *(const v2f*)(wt1 + ko);
            acc0 = __builtin_amdgcn_wmma_f32_16x16x4_f32(
                false, a, false, b0, (short)0, acc0, false, false);
            acc1 = __builtin_amdgcn_wmma_f32_16x16x4_f32(
                false, a, false, b1, (short)0, acc1, false, false);
        }

        // ---- epilogue: residual + scatter into split (:,9)/(:,13) layout ----
        const size_t n4tot = (size_t)Npix * 9;
        #pragma unroll
        for (int r = 0; r < 8; ++r) {
            int m = r + (hiHalf ? 8 : 0);            // pixel row in tile
            size_t gp = (size_t)tile * 16 + (size_t)m;
            if (gp < (size_t)Npix) {
                int k0 = laneLo;                      // 0..15
                float v0 = acc0[r] + sXY[wave][m][k0];
                if (k0 < 9) out[gp * 9 + k0] = v0;
                else        out[n4tot + gp * 13 + (k0 - 9)] = v0;
                int k1 = 16 + laneLo;                 // 16..31
                if (k1 < 22) {
                    float v1 = acc1[r] + sXY[wave][m][k1];
                    out[n4tot + gp * 13 + (k1 - 9)] = v1;
                }
            }
        }
    }
}

// =====================================================================
extern "C" void kernel_launch(void* const* d_in, const int* in_sizes, int n_in,
                              void* d_out, int out_size, void* d_ws, size_t ws_size,
                              hipStream_t stream) {
    const float* f4  = (const float*)d_in[0];
    const float* f6  = (const float*)d_in[1];
    const float* sw  = (const float*)d_in[2];
    const float* tpw = (const float*)d_in[3];
    const int*   pH  = (const int*)d_in[4];
    const int*   pW  = (const int*)d_in[5];
    float* out = (float*)d_out;

    const int Npix = in_sizes[0] / 9;

    float* ws     = (float*)d_ws;
    float* d_Wt   = ws;                   // 15616 floats (W^T padded 32x488)
    float* d_base = ws + NWELEM;          // 10648 floats (raw 3j tensors)
    float* d_norm = d_base + 10648;       // 8 floats (per-path Frobenius^2)

    esc_zero<<<1, 32, 0, stream>>>(d_norm);
    esc_wigner<<<(10648 + 255) / 256, 256, 0, stream>>>(d_base, d_norm);
    esc_combine<<<(32 * NWCOL + 255) / 256, 256, 0, stream>>>(tpw, d_base, d_norm, d_Wt);

    const int nTiles = (Npix + 15) / 16;
    const int nWG    = (nTiles + WAVES_PER_WG - 1) / WAVES_PER_WG;
    esc_main<<<nWG, 256, 0, stream>>>(f4, f6, sw, pH, pW, d_Wt, out, Npix, nTiles);
}